// EdgeEncoder_12927851561320
// MI455X (gfx1250) — compile-verified
//
#include <hip/hip_runtime.h>

typedef __attribute__((ext_vector_type(2))) float v2f;
typedef __attribute__((ext_vector_type(8))) float v8f;

#define D_FEAT 128
#define LDS_STRIDE 33   // odd stride: conflict-free on 64 banks, room for 32 cols

static __device__ __forceinline__ v8f wmma4(v2f a, v2f b, v8f c) {
    // D(16x16,f32) = A(16x4,f32) * B(4x16,f32) + C
    return __builtin_amdgcn_wmma_f32_16x16x4_f32(
        /*neg_a=*/false, a, /*neg_b=*/false, b,
        /*c_mod=*/(short)0, c, /*reuse_a=*/false, /*reuse_b=*/false);
}

__global__ __launch_bounds__(256) void edge_encoder_kernel(
    const float* __restrict__ x,
    const long long* __restrict__ edge_index,
    const float* __restrict__ edge_attr,
    const float* __restrict__ W1, const float* __restrict__ b1,
    const float* __restrict__ W2, const float* __restrict__ b2,
    float* __restrict__ out,
    int E, int ntiles)
{
    __shared__ float hb[8][16 * LDS_STRIDE];   // per-wave h tile (16x18 used)

    const int lane = threadIdx.x & 31;
    const int wid  = threadIdx.x >> 5;
    const int m    = lane & 15;     // row (A) / col (B,C) index within tile
    const int half = lane >> 4;     // lane half selects K pair per ISA layout
    const int klo  = half * 2;

    float* hw = hb[wid];

    // ---------------- hoisted weight / bias fragments ----------------
    // Fragment layout (f32 16x16x4): A: lane=M, v.x=K(klo), v.y=K(klo+1)
    //                                B: lane=N, v.x=K(klo), v.y=K(klo+1)
    v2f w1b[2][2];      // [K-chunk][N-half], W1 is 5x18 row-major (K=5 padded to 8)
    v2f w2b[5][2];      // [K-chunk][N-half], W2 is 18x32 row-major (K=18 padded to 20)
    float b1v[2], b2v[2];
    #pragma unroll
    for (int hN = 0; hN < 2; ++hN) {
        const int col = hN * 16 + m;
        const bool cOK = (col < 18);
        w1b[0][hN].x = cOK ? W1[klo * 18 + col] : 0.f;
        w1b[0][hN].y = cOK ? W1[(klo + 1) * 18 + col] : 0.f;
        w1b[1][hN].x = (cOK && half == 0) ? W1[4 * 18 + col] : 0.f;  // K=4 only
        w1b[1][hN].y = 0.f;                                          // K=5,7 pad
        b1v[hN] = cOK ? b1[col] : 0.f;
        b2v[hN] = b2[hN * 16 + m];
        #pragma unroll
        for (int c = 0; c < 5; ++c) {
            const int k0 = c * 4 + klo;
            w2b[c][hN].x = (k0     < 18) ? W2[k0 * 32 + hN * 16 + m] : 0.f;
            w2b[c][hN].y = (k0 + 1 < 18) ? W2[(k0 + 1) * 32 + hN * 16 + m] : 0.f;
        }
    }

    const long long* srcI = edge_index;       // edge_index[0][*]
    const long long* tgtI = edge_index + E;   // edge_index[1][*]
    const int wavesTotal = gridDim.x * 8;

    for (int tile = blockIdx.x * 8 + wid; tile < ntiles; tile += wavesTotal) {
        const int base = tile * 16;

        // ---- phase A: cosine similarity, 2 edges/iter (16-lane groups) ----
        float cosKeep = 0.f;   // lane e ends up holding cos(edge base+e), e<16
        for (int j = 0; j < 8; ++j) {
            const int e0 = base + 2 * j;
            if (e0 >= E) break;                      // uniform branch
            const int  ei = e0 + half;               // group's edge
            const bool ok = (ei < E);
            float4 slo = {0,0,0,0}, shi = {0,0,0,0};
            float4 tlo = {0,0,0,0}, thi = {0,0,0,0};
            if (ok) {
                const long long si = srcI[ei];
                const long long ti = tgtI[ei];
                const float4* sr = (const float4*)(x + si * (long long)D_FEAT);
                const float4* tr = (const float4*)(x + ti * (long long)D_FEAT);
                slo = sr[m]; shi = sr[m + 16];       // 16 lanes x 16B = 256B runs
                tlo = tr[m]; thi = tr[m + 16];
            }
            float d  = slo.x*tlo.x + slo.y*tlo.y + slo.z*tlo.z + slo.w*tlo.w
                     + shi.x*thi.x + shi.y*thi.y + shi.z*thi.z + shi.w*thi.w;
            float na = slo.x*slo.x + slo.y*slo.y + slo.z*slo.z + slo.w*slo.w
                     + shi.x*shi.x + shi.y*shi.y + shi.z*shi.z + shi.w*shi.w;
            float nb = tlo.x*tlo.x + tlo.y*tlo.y + tlo.z*tlo.z + tlo.w*tlo.w
                     + thi.x*thi.x + thi.y*thi.y + thi.z*thi.z + thi.w*thi.w;
            #pragma unroll
            for (int off = 8; off > 0; off >>= 1) {  // stays within 16-lane group
                d  += __shfl_xor(d,  off, 32);
                na += __shfl_xor(na, off, 32);
                nb += __shfl_xor(nb, off, 32);
            }
            const float ns = fmaxf(sqrtf(na), 1e-8f);
            const float nt = fmaxf(sqrtf(nb), 1e-8f);
            const float cv  = d / (ns * nt);         // this group's cos
            const float ocv = __shfl_xor(cv, 16, 32);// other group's cos
            if (lane == 2 * j)     cosKeep = cv;     // edge 2j  (group 0 value)
            if (lane == 2 * j + 1) cosKeep = ocv;    // edge 2j+1 (group 1 value)
        }

        // ---------- layer 1: feats[16x5] @ W1[5x18] via 4 WMMAs ----------
        v2f a0; a0.x = 0.f; a0.y = 0.f;              // K-chunk 0 = edge_attr
        {
            const int ea = base + m;
            if (ea < E) {
                const float2 t = *(const float2*)(edge_attr + (long long)ea * 4 + klo);
                a0.x = t.x; a0.y = t.y;
            }
        }
        v2f a1;                                      // K-chunk 1: K=4 is cos
        a1.x = (half == 0) ? cosKeep : 0.f;
        a1.y = 0.f;

        v8f c0 = {}; v8f c1 = {};
        c0 = wmma4(a0, w1b[0][0], c0);
        c0 = wmma4(a1, w1b[1][0], c0);
        c1 = wmma4(a0, w1b[0][1], c1);
        c1 = wmma4(a1, w1b[1][1], c1);

        // bias + relu, spill h through per-wave LDS (C-layout -> A-layout).
        // Unconditional stores (cols 18..31 are scratch, never read back).
        #pragma unroll
        for (int r = 0; r < 8; ++r) {
            const int row = r + half * 8;
            hw[row * LDS_STRIDE + m]      = fmaxf(c0[r] + b1v[0], 0.f);
            hw[row * LDS_STRIDE + 16 + m] = fmaxf(c1[r] + b1v[1], 0.f);
        }

        // ---------- layer 2: h[16x18] @ W2[18x32] via 10 WMMAs ----------
        v8f d0 = {}; v8f d1 = {};
        #pragma unroll
        for (int c = 0; c < 5; ++c) {
            const int k0 = c * 4 + klo;
            v2f a;
            a.x = (k0     < 18) ? hw[m * LDS_STRIDE + k0]     : 0.f;
            a.y = (k0 + 1 < 18) ? hw[m * LDS_STRIDE + k0 + 1] : 0.f;
            d0 = wmma4(a, w2b[c][0], d0);
            d1 = wmma4(a, w2b[c][1], d1);
        }

        // ---------- bias + relu + store [E,32] ----------
        #pragma unroll
        for (int r = 0; r < 8; ++r) {
            const int erow = base + r + half * 8;
            if (erow < E) {
                out[(long long)erow * 32 + m]      = fmaxf(d0[r] + b2v[0], 0.f);
                out[(long long)erow * 32 + 16 + m] = fmaxf(d1[r] + b2v[1], 0.f);
            }
        }
    }
}

extern "C" void kernel_launch(void* const* d_in, const int* in_sizes, int n_in,
                              void* d_out, int out_size, void* d_ws, size_t ws_size,
                              hipStream_t stream) {
    const float*     x          = (const float*)d_in[0];
    const long long* edge_index = (const long long*)d_in[1];  // int64 [2,E]
    const float*     edge_attr  = (const float*)d_in[2];
    const float*     W1         = (const float*)d_in[3];
    const float*     b1         = (const float*)d_in[4];
    const float*     W2         = (const float*)d_in[5];
    const float*     b2         = (const float*)d_in[6];
    float*           out        = (float*)d_out;

    const int E      = in_sizes[1] / 2;
    const int ntiles = (E + 15) / 16;
    int blocks = (ntiles + 7) / 8;       // one wave (32 lanes) per 16-edge tile
    if (blocks < 1) blocks = 1;

    edge_encoder_kernel<<<blocks, 256, 0, stream>>>(
        x, edge_index, edge_attr, W1, b1, W2, b2, out, E, ntiles);
}